// DiffConv_81106162417816
// MI455X (gfx1250) — compile-verified
//
#include <hip/hip_runtime.h>

// ---------------------------------------------------------------------------
// DiffConv on MI455X (gfx1250), fp32 end-to-end via V_WMMA_F32_16X16X4_F32.
// HBM-bound (~400 MB traffic vs ~0.27 TFLOP at 23.3 TB/s => ~17 us floor), so
// fp32 WMMA is exact and free.  Data movement uses the CDNA5 async
// global->LDS path (GLOBAL_LOAD_ASYNC_TO_LDS_B128 + s_wait_asynccnt) with
// double-buffered tiles so each tile fetch overlaps a WMMA burst.
//   Kernel 1: A2[t] = A[t] @ A[t]            (into d_ws, 96 MiB)
//   Kernel 2: out = H(Wf0+Wb0) + A(H Wf1) + A^T(H Wb1)
//                  + A2(H Wf2) + A2^T(H Wb2) + bias      (fused)
// ---------------------------------------------------------------------------

typedef __attribute__((ext_vector_type(2))) float v2f;
typedef __attribute__((ext_vector_type(8))) float v8f;
typedef int b128i __attribute__((vector_size(16)));  // builtin's pointee type

#define NN 1024
#define DD 64
#define KSTEPS 24
#define BATCH 8
#define TILE_DIM 64
#define LDSP 68                   // padded LDS row stride (floats): conflict-free
#define TILE_F (TILE_DIM * LDSP)  // 4352 floats per 64x64 padded tile

#if defined(__has_builtin)
#if __has_builtin(__builtin_amdgcn_global_load_async_to_lds_b128)
#define HAS_ASYNC_LDS 1
#endif
#endif
#ifndef HAS_ASYNC_LDS
#define HAS_ASYNC_LDS 0
#endif

__device__ __forceinline__ v8f wmma_f32(v2f a, v2f b, v8f c) {
  // D = A(16x4) * B(4x16) + C(16x16), fp32, wave32
  return __builtin_amdgcn_wmma_f32_16x16x4_f32(false, a, false, b, (short)0, c,
                                               false, false);
}

// Wait for this wave's outstanding async global<->LDS ops to drain.
template <int N>
__device__ __forceinline__ void wait_async() {
#if HAS_ASYNC_LDS
#if __has_builtin(__builtin_amdgcn_s_wait_asynccnt)
  __builtin_amdgcn_s_wait_asynccnt(N);
#else
  asm volatile("s_wait_asynccnt %0" ::"n"(N) : "memory");
#endif
#endif
}

// 64x64 fp32 global tile (row stride gs) -> padded LDS tile, issued by all
// 256 threads (4 x b128 per thread).  Async when available (no VGPR bounce,
// tracked by ASYNCcnt); sync float4 fallback otherwise.
__device__ __forceinline__ void tile_load(float* dst, const float* src,
                                          int gs) {
  const int r = threadIdx.x >> 2;            // 0..63
  const int cbase = (threadIdx.x & 3) << 4;  // 0,16,32,48
#pragma unroll
  for (int u = 0; u < 4; ++u) {
    const int c = cbase + (u << 2);
#if HAS_ASYNC_LDS
    __builtin_amdgcn_global_load_async_to_lds_b128(
        (__attribute__((address_space(1))) b128i*)(src + (size_t)r * gs + c),
        (__attribute__((address_space(3))) b128i*)(dst + r * LDSP + c), 0, 0);
#else
    *(float4*)(dst + r * LDSP + c) = *(const float4*)(src + (size_t)r * gs + c);
#endif
  }
}

// --- fragment loaders (layouts per CDNA5 ISA 7.12.2) -----------------------
// A 16x4: lanes 0-15 -> rows, K pair 0/1; lanes 16-31 -> rows, K pair 2/3
__device__ __forceinline__ v2f lds_afrag(const float* s, int rbase, int kbase) {
  const int lane = threadIdx.x & 31;
  const int r = rbase + (lane & 15);
  const int k = kbase + ((lane >> 4) << 1);
  v2f a;
  a.x = s[r * LDSP + k];
  a.y = s[r * LDSP + k + 1];
  return a;
}

// Same logical A-frag, but LDS tile holds the transpose: logical[i][k]=s[k][i]
__device__ __forceinline__ v2f lds_afragT(const float* s, int rbase,
                                          int kbase) {
  const int lane = threadIdx.x & 31;
  const int r = rbase + (lane & 15);
  const int k = kbase + ((lane >> 4) << 1);
  v2f a;
  a.x = s[k * LDSP + r];
  a.y = s[(k + 1) * LDSP + r];
  return a;
}

// B 4x16: VGPR0 = rows K0 (lanes 0-15) / K2 (lanes 16-31); VGPR1 = K1/K3
__device__ __forceinline__ v2f lds_bfrag(const float* s, int kbase, int cbase) {
  const int lane = threadIdx.x & 31;
  const int c = cbase + (lane & 15);
  const int k = kbase + ((lane >> 4) << 1);
  v2f b;
  b.x = s[k * LDSP + c];
  b.y = s[(k + 1) * LDSP + c];
  return b;
}

// B-frag of a 64x64 row-major matrix straight from global (W: 48 KB, L1-hot)
__device__ __forceinline__ v2f g_bfrag(const float* w, int kbase, int cbase) {
  const int lane = threadIdx.x & 31;
  const int c = cbase + (lane & 15);
  const int k = kbase + ((lane >> 4) << 1);
  v2f b;
  b.x = w[k * DD + c];
  b.y = w[(k + 1) * DD + c];
  return b;
}

// C/D 16x16: VGPR v holds row v (lanes 0-15) / row v+8 (lanes 16-31)
__device__ __forceinline__ void lds_store_ctile(float* s, int rbase, int cbase,
                                                v8f c) {
  const int lane = threadIdx.x & 31;
  const int col = cbase + (lane & 15);
  const int r = rbase + ((lane >> 4) << 3);
#pragma unroll
  for (int v = 0; v < 8; ++v) s[(r + v) * LDSP + col] = c[v];
}

__device__ __forceinline__ void g_store_ctile(float* g, int gstride, int rbase,
                                              int cbase, v8f c, float add) {
  const int lane = threadIdx.x & 31;
  const int col = cbase + (lane & 15);
  const int r = rbase + ((lane >> 4) << 3);
#pragma unroll
  for (int v = 0; v < 8; ++v) g[(size_t)(r + v) * gstride + col] = c[v] + add;
}

// ---------------------------------------------------------------------------
// Kernel 1: A2[t] = A[t] @ A[t]   (24 x 1024x1024 fp32 GEMMs)
// Double-buffered async tiles: issue j+1, compute j, drain, barrier.
// ---------------------------------------------------------------------------
__global__ __launch_bounds__(256) void a2_kernel(const float* __restrict__ A,
                                                 float* __restrict__ A2) {
  __shared__ __align__(16) float sRow[2][TILE_F];
  __shared__ __align__(16) float sCol[2][TILE_F];
  const int t = blockIdx.z;
  const int r0 = blockIdx.y * TILE_DIM;
  const int c0 = blockIdx.x * TILE_DIM;
  const float* At = A + (size_t)t * NN * NN;
  float* A2t = A2 + (size_t)t * NN * NN;

  const int wave = threadIdx.x >> 5;   // 8 waves
  const int sub_r = (wave >> 1) * 16;  // 4 row subtiles
  const int sub_c = (wave & 1) * 32;   // 2 col-pair positions

  v8f acc0 = {};
  v8f acc1 = {};

  tile_load(sRow[0], At + (size_t)r0 * NN, NN);
  tile_load(sCol[0], At + c0, NN);
  wait_async<0>();
  __syncthreads();

  for (int jj = 0; jj < 16; ++jj) {
    const int buf = jj & 1;
    if (jj + 1 < 16) {  // prefetch next K-chunk while computing this one
      const int j2 = (jj + 1) * TILE_DIM;
      tile_load(sRow[buf ^ 1], At + (size_t)r0 * NN + j2, NN);
      tile_load(sCol[buf ^ 1], At + (size_t)j2 * NN + c0, NN);
    }
#pragma unroll
    for (int kk = 0; kk < 16; ++kk) {
      const v2f a = lds_afrag(sRow[buf], sub_r, kk * 4);
      const v2f b0 = lds_bfrag(sCol[buf], kk * 4, sub_c);
      const v2f b1 = lds_bfrag(sCol[buf], kk * 4, sub_c + 16);
      acc0 = wmma_f32(a, b0, acc0);
      acc1 = wmma_f32(a, b1, acc1);
    }
    wait_async<0>();
    __syncthreads();
  }
  g_store_ctile(A2t, NN, r0 + sub_r, c0 + sub_c, acc0, 0.0f);
  g_store_ctile(A2t, NN, r0 + sub_r, c0 + sub_c + 16, acc1, 0.0f);
}

// ---------------------------------------------------------------------------
// Kernel 2: fused diffusion conv. One block per (64-row tile, b, t).
// Per term m = 4c+term: issue async tile for m+1, stage-1 WMMAs (Y = Hc @ W),
// drain+barrier, stage-2 WMMAs (acc += A_tile @ Y), barrier.
// ---------------------------------------------------------------------------
__global__ __launch_bounds__(256) void diffconv_kernel(
    const float* __restrict__ H, const float* __restrict__ A,
    const float* __restrict__ A2, const float* __restrict__ Wf,
    const float* __restrict__ Wb, const float* __restrict__ bias,
    float* __restrict__ out) {
  __shared__ __align__(16) float sH[2][TILE_F];  // H K-chunks (double buffer)
  __shared__ __align__(16) float sY[TILE_F];     // Y = Hchunk @ W (one term)
  __shared__ __align__(16) float sA[2][TILE_F];  // A/A2 tiles (double buffer)

  const int r0 = blockIdx.x * TILE_DIM;
  const int b = blockIdx.y;
  const int t = blockIdx.z;

  const float* Ht = H + (size_t)(b * KSTEPS + t) * NN * DD;
  const float* At = A + (size_t)t * NN * NN;
  const float* A2t = A2 + (size_t)t * NN * NN;
  float* Ot = out + (size_t)(b * KSTEPS + t) * NN * DD;

  const int wave = threadIdx.x >> 5;
  const int c_sub_r = (wave >> 1) * 16;  // this wave's C rows within tile
  const int c_sub_c = (wave & 1) * 32;   // this wave's two 16-wide C cols
  const int y_sub0 = wave * 2;           // this wave's two Y subtiles (of 16)

  v8f acc0 = {};
  v8f acc1 = {};

  // A/A2 tile pointer for term nt of K-chunk starting at row/col nj
  auto tile_ptr = [&](int nt, int nj) -> const float* {
    return (nt == 0)   ? (At + (size_t)r0 * NN + nj)    // A    row tile
           : (nt == 1) ? (At + (size_t)nj * NN + r0)    // A^T  (col tile)
           : (nt == 2) ? (A2t + (size_t)r0 * NN + nj)   // A2   row tile
                       : (A2t + (size_t)nj * NN + r0);  // A2^T (col tile)
  };

  // ---- prologue: async-stage H[r0] (into sY), first A tile, first H chunk
  tile_load(sY, Ht + (size_t)r0 * DD, DD);
  tile_load(sA[0], tile_ptr(0, 0), NN);
  tile_load(sH[0], Ht, DD);
  wait_async<0>();
  __syncthreads();

  // direct term: H_tile @ (Wf0 + Wb0)
#pragma unroll
  for (int kk = 0; kk < 16; ++kk) {
    const v2f a = lds_afrag(sY, c_sub_r, kk * 4);
    const v2f f0 = g_bfrag(Wf, kk * 4, c_sub_c);
    const v2f g0 = g_bfrag(Wb, kk * 4, c_sub_c);
    const v2f f1 = g_bfrag(Wf, kk * 4, c_sub_c + 16);
    const v2f g1 = g_bfrag(Wb, kk * 4, c_sub_c + 16);
    v2f b0, b1;
    b0.x = f0.x + g0.x;
    b0.y = f0.y + g0.y;
    b1.x = f1.x + g1.x;
    b1.y = f1.y + g1.y;
    acc0 = wmma_f32(a, b0, acc0);
    acc1 = wmma_f32(a, b1, acc1);
  }
  __syncthreads();  // sY free for term-0 stage-1 writes

  // ---- main loop over K chunks of the N=1024 reduction ----
  for (int c = 0; c < 16; ++c) {
    const int j = c * TILE_DIM;
    const float* sHc = sH[c & 1];
#pragma unroll
    for (int term = 0; term < 4; ++term) {
      // term 0: A    @ (H Wf1)    term 1: A^T  @ (H Wb1)
      // term 2: A2   @ (H Wf2)    term 3: A2^T @ (H Wb2)
      const int m = c * 4 + term;
      const float* Wsel = (term == 0)   ? (Wf + DD * DD)
                          : (term == 1) ? (Wb + DD * DD)
                          : (term == 2) ? (Wf + 2 * DD * DD)
                                        : (Wb + 2 * DD * DD);

      // prefetch the NEXT term's A/A2 tile (and next H chunk at term 3);
      // overlaps the stage-1 WMMA burst below.
      if (m + 1 < 64) {
        const int nt = (term + 1) & 3;
        const int nj = (term == 3) ? j + TILE_DIM : j;
        tile_load(sA[(m + 1) & 1], tile_ptr(nt, nj), NN);
      }
      if (term == 3 && c + 1 < 16) {
        tile_load(sH[(c + 1) & 1], Ht + (size_t)(j + TILE_DIM) * DD, DD);
      }

      // stage 1: Y = Hchunk @ Wsel  (64x64x64, split 2 subtiles/wave)
#pragma unroll
      for (int s = 0; s < 2; ++s) {
        const int sub = y_sub0 + s;
        const int yr = (sub >> 2) * 16;
        const int yc = (sub & 3) * 16;
        v8f y = {};
#pragma unroll
        for (int kk = 0; kk < 16; ++kk) {
          const v2f a = lds_afrag(sHc, yr, kk * 4);
          const v2f w = g_bfrag(Wsel, kk * 4, yc);
          y = wmma_f32(a, w, y);
        }
        lds_store_ctile(sY, yr, yc, y);
      }
      wait_async<0>();  // this wave's async tiles (incl. sA[m]) landed
      __syncthreads();  // sY + all waves' tile portions visible

      // stage 2: acc += Atile (or Atile^T) @ Y
#pragma unroll
      for (int kk = 0; kk < 16; ++kk) {
        const v2f a = (term & 1) ? lds_afragT(sA[m & 1], c_sub_r, kk * 4)
                                 : lds_afrag(sA[m & 1], c_sub_r, kk * 4);
        const v2f b0 = lds_bfrag(sY, kk * 4, c_sub_c);
        const v2f b1 = lds_bfrag(sY, kk * 4, c_sub_c + 16);
        acc0 = wmma_f32(a, b0, acc0);
        acc1 = wmma_f32(a, b1, acc1);
      }
      __syncthreads();  // buffers free for next term's writes
    }
  }

  // ---- epilogue: + bias, store ----
  const int lane = threadIdx.x & 31;
  const float bias0 = bias[c_sub_c + (lane & 15)];
  const float bias1 = bias[c_sub_c + 16 + (lane & 15)];
  g_store_ctile(Ot, DD, r0 + c_sub_r, c_sub_c, acc0, bias0);
  g_store_ctile(Ot, DD, r0 + c_sub_r, c_sub_c + 16, acc1, bias1);
}

// ---------------------------------------------------------------------------
extern "C" void kernel_launch(void* const* d_in, const int* in_sizes, int n_in,
                              void* d_out, int out_size, void* d_ws,
                              size_t ws_size, hipStream_t stream) {
  const float* H = (const float*)d_in[0];     // [8,24,1024,64]
  const float* A = (const float*)d_in[1];     // [24,1024,1024]
  const float* Wf = (const float*)d_in[2];    // [3,64,64]
  const float* Wb = (const float*)d_in[3];    // [3,64,64]
  const float* bias = (const float*)d_in[4];  // [64]
  float* out = (float*)d_out;                 // [8,24,1024,64]
  float* A2 = (float*)d_ws;                   // 24*1024*1024 fp32 = 96 MiB

  dim3 blk(256, 1, 1);
  a2_kernel<<<dim3(NN / TILE_DIM, NN / TILE_DIM, KSTEPS), blk, 0, stream>>>(A,
                                                                           A2);
  // grid ordered (row-tile, b, t): all 128 blocks of one t run adjacently so
  // A[t]/A2[t] (8 MB) stay hot in the 192 MB L2 across the batch dimension.
  diffconv_kernel<<<dim3(NN / TILE_DIM, BATCH, KSTEPS), blk, 0, stream>>>(
      H, A, A2, Wf, Wb, bias, out);

  (void)in_sizes;
  (void)n_in;
  (void)out_size;
  (void)ws_size;
}